// FrageEmbedding_26302379720976
// MI455X (gfx1250) — compile-verified
//
#include <hip/hip_runtime.h>

// FrageEmbedding for MI455X (gfx1250, wave32).
// Bandwidth-bound (~34 MB HBM traffic); compute runs on V_WMMA_F32_16X16X4_F32
// after binning tokens by partition so each 16-token tile is a dense f32 GEMM
// against a single (L2-resident) weight matrix.

typedef __attribute__((ext_vector_type(2))) float v2f;
typedef __attribute__((ext_vector_type(8))) float v8f;

#define NTOK 16384   // B*S = 16*1024
#define DDIM 256     // D
#define NPART 8      // P

__global__ void frage_zero_cnt(int* __restrict__ cnt) {
    if (threadIdx.x < NPART) cnt[threadIdx.x] = 0;
}

// Phase 1: bin flat token positions by partition id.
__global__ void frage_bin(const int* __restrict__ x,
                          const int* __restrict__ partitions,
                          int* __restrict__ cnt,
                          int* __restrict__ bucket) {
    int i = blockIdx.x * blockDim.x + threadIdx.x;
    if (i < NTOK) {
        int tok = x[i];
        int p   = partitions[tok];
        int pos = atomicAdd(&cnt[p], 1);
        bucket[p * NTOK + pos] = i;
    }
}

// Phase 2: one block = (partition p, tile of 16 tokens of that partition).
// 8 waves; wave w computes N-tiles 2w, 2w+1 of the 16x256 output tile.
// K loop covers only the active dims (256 >> p); A columns >= active zeroed.
__global__ __launch_bounds__(256) void frage_gemm(
    const int*   __restrict__ x,
    const float* __restrict__ emb_table,
    const float* __restrict__ weight,
    const int*   __restrict__ cnt,
    const int*   __restrict__ bucket,
    float*       __restrict__ out) {

    constexpr int LDA = DDIM + 2;       // even pad: 8B-aligned float2, spreads LDS banks
    __shared__ float As[16 * LDA];      // 16 tokens x 256 dims (f32)
    __shared__ int   toks[16];          // flat token positions (-1 = pad row)

    const int p         = blockIdx.y;
    const int count     = cnt[p];
    const int tileStart = blockIdx.x * 16;
    if (tileStart >= count) return;     // uniform per block -> EXEC stays full
    const int nTok = min(16, count - tileStart);
    const int tid  = threadIdx.x;

    if (tid < 16)
        toks[tid] = (tid < nTok) ? bucket[p * NTOK + tileStart + tid] : -1;
    __syncthreads();

    const int active = DDIM >> p;

    // Stage A: thread (r = tid>>4) loads 16 consecutive dims of token row r.
    {
        const int r  = tid >> 4;
        const int cb = (tid & 15) << 4;
        const int fi = toks[r];
        if (fi >= 0) {
            const float* src = emb_table + (size_t)x[fi] * DDIM;
            #pragma unroll
            for (int j = 0; j < 16; ++j) {
                int c = cb + j;
                As[r * LDA + c] = (c < active) ? src[c] : 0.0f;  // fold the mask into A
            }
        } else {
            #pragma unroll
            for (int j = 0; j < 16; ++j) As[r * LDA + cb + j] = 0.0f;
        }
    }
    __syncthreads();

    const int wave = tid >> 5;
    const int lane = tid & 31;
    const int half = lane >> 4;   // 0: lanes 0-15, 1: lanes 16-31
    const int l15  = lane & 15;
    const int kIters = (active + 3) >> 2;   // p=7 -> active=2 -> 1 iter (cols 2,3 are zero in A)

    const float* Wp    = weight + (size_t)p * DDIM * DDIM;
    const int    zb0   = (wave * 2) * 16;
    const int    zb1   = zb0 + 16;
    const float* wrow0 = Wp + (size_t)(zb0 + l15) * DDIM;   // B[k][n] = weight[p][zb+n][k]
    const float* wrow1 = Wp + (size_t)(zb1 + l15) * DDIM;

    v8f acc0 = {};
    v8f acc1 = {};

    for (int kk = 0; kk < kIters; ++kk) {
        // A-frag layout (ISA 16x4 f32): lanes 0-15 hold K=k,k+1; lanes 16-31 hold K=k+2,k+3.
        // B-frag mirrored (4x16): per-lane pair at the same K offsets, N = lane&15.
        const int k = (kk << 2) + (half << 1);
        v2f a  = *(const v2f*)&As[l15 * LDA + k];   // ds_load_b64
        v2f b0 = *(const v2f*)&wrow0[k];            // global_load_b64 (L2 hit: weights 2MB)
        v2f b1 = *(const v2f*)&wrow1[k];
        acc0 = __builtin_amdgcn_wmma_f32_16x16x4_f32(false, a, false, b0,
                                                     (short)0, acc0, false, false);
        acc1 = __builtin_amdgcn_wmma_f32_16x16x4_f32(false, a, false, b1,
                                                     (short)0, acc1, false, false);
    }

    // C/D layout: VGPR j, lanes 0-15 -> (M=j, N=l15); lanes 16-31 -> (M=j+8, N=l15).
    #pragma unroll
    for (int j = 0; j < 8; ++j) {
        const int m  = j + (half << 3);
        const int fi = toks[m];
        if (fi >= 0) {   // skip pad rows of tail tiles
            out[(size_t)fi * DDIM + zb0 + l15] = acc0[j];
            out[(size_t)fi * DDIM + zb1 + l15] = acc1[j];
        }
    }
}

extern "C" void kernel_launch(void* const* d_in, const int* in_sizes, int n_in,
                              void* d_out, int out_size, void* d_ws, size_t ws_size,
                              hipStream_t stream) {
    const int*   x          = (const int*)d_in[0];     // [B,S] int32
    const float* emb_table  = (const float*)d_in[1];   // [V,D] f32
    const float* weight     = (const float*)d_in[2];   // [P,D,D] f32
    const int*   partitions = (const int*)d_in[3];     // [V] int32
    float*       out        = (float*)d_out;           // [B,S,D] f32

    // Workspace: cnt[8] at offset 0 (padded to 256B), buckets[8][16384] after.
    int* cnt    = (int*)d_ws;
    int* bucket = (int*)((char*)d_ws + 256);

    frage_zero_cnt<<<1, 32, 0, stream>>>(cnt);
    frage_bin<<<(NTOK + 255) / 256, 256, 0, stream>>>(x, partitions, cnt, bucket);

    // Worst case: all tokens in one partition -> 1024 tiles; empty tiles exit early.
    dim3 grid(NTOK / 16, NPART);
    frage_gemm<<<grid, 256, 0, stream>>>(x, emb_table, weight, cnt, bucket, out);
}